// Linear_16320875725432
// MI455X (gfx1250) — compile-verified
//
#include <hip/hip_runtime.h>

// Problem constants (match reference)
#define IN_F   256
#define OUT_F  128
#define TABLES (IN_F * OUT_F)   // 32768
#define BATCH  256

typedef __attribute__((ext_vector_type(2))) float v2f;
typedef __attribute__((ext_vector_type(8))) float v8f;

// ---------------------------------------------------------------------------
// Kernel 0: zero the scattered weight matrix, seed Abias with the bias vector.
// ---------------------------------------------------------------------------
__global__ void init_ws(float* __restrict__ Wd, float* __restrict__ Abias,
                        const float* __restrict__ bias) {
    int i = blockIdx.x * blockDim.x + threadIdx.x;
    if (i < IN_F * OUT_F) Wd[i] = 0.0f;
    if (i < OUT_F)        Abias[i] = bias[i];
}

// ---------------------------------------------------------------------------
// Kernel 1: per-table Lagrange -> multilinear coefficients; scatter the linear
// coefficients into a dense 256x128 matrix Wd (row-major, [j][o]) and reduce
// the constant coefficients into Abias[o].
// ---------------------------------------------------------------------------
__global__ void scatter_coeffs(const float* __restrict__ w,
                               const int*   __restrict__ mask,
                               float* __restrict__ Wd,
                               float* __restrict__ Abias) {
    int t = blockIdx.x * blockDim.x + threadIdx.x;
    if (t >= TABLES) return;
    float w0 = w[4*t+0], w1 = w[4*t+1], w2 = w[4*t+2], w3 = w[4*t+3];
    float a  = 0.25f * ( w0 + w1 + w2 + w3);
    float b0 = 0.25f * (-w0 + w1 - w2 + w3);
    float b1 = 0.25f * (-w0 - w1 + w2 + w3);
    int o  = t >> 8;               // t = o*256 + i
    int m0 = mask[2*t + 0];
    int m1 = mask[2*t + 1];
    atomicAdd(&Wd[m0 * OUT_F + o], b0);
    atomicAdd(&Wd[m1 * OUT_F + o], b1);
    atomicAdd(&Abias[o], a);
}

// ---------------------------------------------------------------------------
// Kernel 2: dense GEMM  out[b][o] = sum_j X[b][j] * Wd[j][o] + Abias[o]
// via V_WMMA_F32_16X16X4_F32. One wave per 16x16 output tile.
// Grid: 16 blocks x 256 threads (8 waves) covering 16x8 = 128 tiles.
// ---------------------------------------------------------------------------
__global__ void __launch_bounds__(256)
wmma_gemm(const float* __restrict__ X,     // [BATCH][IN_F]
          const float* __restrict__ Wd,    // [IN_F][OUT_F]
          const float* __restrict__ Abias, // [OUT_F]
          float* __restrict__ out) {       // [BATCH][OUT_F]
    const int wave  = threadIdx.x >> 5;
    const int lane  = threadIdx.x & 31;
    const int tile  = blockIdx.x * 8 + wave;   // 0..127
    const int mtile = tile >> 3;               // 0..15 (batch tiles)
    const int ntile = tile & 7;                // 0..7  (output tiles)

    const int m     = lane & 15;   // row within A fragment / col within B,C
    const int khalf = lane >> 4;   // selects K pair {0,1} vs {2,3}

    const int rowA = mtile * 16 + m;          // batch row this lane loads
    const int colB = ntile * 16 + m;          // output col this lane loads

    v8f acc = {};
    for (int kb = 0; kb < IN_F; kb += 4) {
        // A 16x4 f32: VGPR v holds K = 2*khalf + v for lane's row (ISA 7.12.2)
        const float* ap = X + rowA * IN_F + kb + 2 * khalf;
        v2f a; a.x = ap[0]; a.y = ap[1];
        // B 4x16 f32: VGPR v holds row K = 2*khalf + v, column = lane&15
        const float* bp = Wd + (kb + 2 * khalf) * OUT_F + colB;
        v2f b; b.x = bp[0]; b.y = bp[OUT_F];
        acc = __builtin_amdgcn_wmma_f32_16x16x4_f32(
            /*neg_a=*/false, a, /*neg_b=*/false, b,
            /*c_mod=*/(short)0, acc, /*reuse_a=*/false, /*reuse_b=*/false);
    }

    // C/D layout: VGPR r -> M = r + 8*khalf, N = lane&15
    const float ab = Abias[colB];
#pragma unroll
    for (int r = 0; r < 8; ++r) {
        int row = mtile * 16 + r + 8 * khalf;
        out[row * OUT_F + colB] = acc[r] + ab;
    }
}

// ---------------------------------------------------------------------------
// Kernel 3: quadratic cross term  out[b][o] += sum_i b01[o,i]*x[m0]*x[m1]
// One block per batch row (input row staged in LDS), one thread per output.
// ---------------------------------------------------------------------------
__global__ void __launch_bounds__(128)
quad_term(const float* __restrict__ X,
          const float* __restrict__ w,
          const int*   __restrict__ mask,
          float* __restrict__ out) {
    __shared__ float xs[IN_F];
    const int b = blockIdx.x;
    const int o = threadIdx.x;
    for (int j = threadIdx.x; j < IN_F; j += blockDim.x)
        xs[j] = X[b * IN_F + j];
    __syncthreads();

    float q = 0.0f;
    const int base = o << 8;   // o * IN_F
#pragma unroll 4
    for (int i = 0; i < IN_F; ++i) {
        const int t = base + i;
        float w0 = w[4*t+0], w1 = w[4*t+1], w2 = w[4*t+2], w3 = w[4*t+3];
        float b01 = 0.25f * (w0 - w1 - w2 + w3);
        int m0 = mask[2*t + 0];
        int m1 = mask[2*t + 1];
        q = fmaf(b01, xs[m0] * xs[m1], q);
    }
    out[b * OUT_F + o] += q;
}

// ---------------------------------------------------------------------------
extern "C" void kernel_launch(void* const* d_in, const int* in_sizes, int n_in,
                              void* d_out, int out_size, void* d_ws, size_t ws_size,
                              hipStream_t stream) {
    const float* X    = (const float*)d_in[0];   // (256, 256) f32
    const float* W    = (const float*)d_in[1];   // (32768, 4) f32
    const float* bias = (const float*)d_in[2];   // (128,)     f32
    const int*   mask = (const int*)d_in[3];     // (65536,)   int

    float* out   = (float*)d_out;                // (256, 128) f32
    float* Wd    = (float*)d_ws;                 // 256*128 f32 scratch
    float* Abias = Wd + IN_F * OUT_F;            // 128 f32 scratch

    init_ws<<<(IN_F * OUT_F + 255) / 256, 256, 0, stream>>>(Wd, Abias, bias);
    scatter_coeffs<<<TABLES / 256, 256, 0, stream>>>(W, mask, Wd, Abias);
    wmma_gemm<<<16, 256, 0, stream>>>(X, Wd, Abias, out);
    quad_term<<<BATCH, 128, 0, stream>>>(X, W, mask, out);
}